// Model_26989574488637
// MI455X (gfx1250) — compile-verified
//
#include <hip/hip_runtime.h>
#include <hip/hip_bf16.h>

// ---------------------------------------------------------------------------
// Model dims
// ---------------------------------------------------------------------------
#define Bsz     16
#define Lseq    1024
#define CIN     21
#define COUT    21
#define DMODEL  512
#define DINNER  1024
#define DSTATE  16
#define DCONV   4
#define DTRANK  32
#define PRED    96
#define NMARK   4
#define KEMB    80          // 63 conv cols + 4 mark cols, zero-padded to 80
#define LKEEP   (Lseq - PRED)   // 928

typedef float v2f __attribute__((ext_vector_type(2)));
typedef float v8f __attribute__((ext_vector_type(8)));

// -ln(10000)/512
#define POSC (-0.0179889460f)

// LDS strides chosen so per-lane pair loads are 8B-aligned and bank-conflict
// free across the 16 active lanes of each half-wave (64 banks, wave32).
#define ASTR 20   // As[row][k], k offset even -> ds_load_b64
#define BSTR 18   // Bs[col][k] (transposed), k offset even -> ds_load_b64

__device__ __forceinline__ float silu_f(float x) {
    return x / (1.0f + __expf(-x));
}

// ---------------------------------------------------------------------------
// RevIN stats: one block per (b, c); mean + std over L
// ---------------------------------------------------------------------------
__global__ __launch_bounds__(256)
void revin_stats(const float* __restrict__ x, float* __restrict__ meanv,
                 float* __restrict__ stdv) {
    int bc = blockIdx.x;                 // 0..335
    int b = bc / CIN, c = bc - b * CIN;
    __shared__ float s1[256];
    __shared__ float s2[256];
    float sum = 0.f, sq = 0.f;
    for (int l = threadIdx.x; l < Lseq; l += 256) {
        float v = x[((size_t)(b * Lseq + l)) * CIN + c];
        sum += v; sq += v * v;
    }
    s1[threadIdx.x] = sum; s2[threadIdx.x] = sq;
    __syncthreads();
    for (int off = 128; off > 0; off >>= 1) {
        if (threadIdx.x < off) {
            s1[threadIdx.x] += s1[threadIdx.x + off];
            s2[threadIdx.x] += s2[threadIdx.x + off];
        }
        __syncthreads();
    }
    if (threadIdx.x == 0) {
        float m = s1[0] * (1.0f / Lseq);
        float var = s2[0] * (1.0f / Lseq) - m * m;
        meanv[bc] = m;
        stdv[bc]  = sqrtf(var + 1e-5f);
    }
}

// ---------------------------------------------------------------------------
// Build stacked embedding weight We[80][512]:
//   rows 0..62 = conv_tok_w ([3,21,512] flat), rows 63..66 = time_w, rest 0
// ---------------------------------------------------------------------------
__global__ __launch_bounds__(256)
void build_we(const float* __restrict__ conv_tok_w,
              const float* __restrict__ time_w, float* __restrict__ We) {
    int e = blockIdx.x * 256 + threadIdx.x;
    if (e >= KEMB * DMODEL) return;
    int r = e >> 9, n = e & (DMODEL - 1);
    float v = 0.f;
    if (r < 63)       v = conv_tok_w[r * DMODEL + n];
    else if (r < 67)  v = time_w[(r - 63) * DMODEL + n];
    We[e] = v;
}

// ---------------------------------------------------------------------------
// Build embedding activation Xe[B*L][80]:
//   cols 0..62 = normalized x at l-1,l,l+1 (circular), 63..66 = x_mark, rest 0
// ---------------------------------------------------------------------------
__global__ __launch_bounds__(256)
void build_xe(const float* __restrict__ x_enc, const float* __restrict__ x_mark,
              const float* __restrict__ meanv, const float* __restrict__ stdv,
              float* __restrict__ Xe) {
    int e = blockIdx.x * 256 + threadIdx.x;
    if (e >= Bsz * Lseq * KEMB) return;
    int row = e / KEMB, col = e - row * KEMB;
    int b = row >> 10, l = row & (Lseq - 1);
    float v = 0.f;
    if (col < 63) {
        int k = col / CIN, c = col - k * CIN;
        int ll = (l + k - 1 + Lseq) & (Lseq - 1);     // circular pad
        int bc = b * CIN + c;
        v = (x_enc[((size_t)((b << 10) + ll)) * CIN + c] - meanv[bc]) / stdv[bc];
    } else if (col < 67) {
        v = x_mark[((size_t)row) * NMARK + (col - 63)];
    }
    Xe[e] = v;
}

// ---------------------------------------------------------------------------
// WMMA f32 GEMM: C[M,N] = A[M,K] @ B[K,N], 64x64 tile, 4 waves/block,
// V_WMMA_F32_16X16X4_F32 (wave32), double-buffered LDS software pipeline.
// B tile stored transposed in LDS so each lane's (K,K+1) pair is one b64 load.
// MODE 0: plain store
// MODE 1: + positional embedding
// MODE 2: in-proj split: col<1024 -> u_raw; col>=1024 & l>=928 -> silu(z)
// MODE 3: + bias[col], softplus (dt path)
// ---------------------------------------------------------------------------
template <int MODE>
__global__ __launch_bounds__(128)
void wmma_gemm(const float* __restrict__ A, const float* __restrict__ B,
               float* __restrict__ C, float* __restrict__ aux,
               const float* __restrict__ bias,
               int M, int N, int K, int lda, int ldb, int ldc) {
    __shared__ float As[2][64 * ASTR];
    __shared__ float Bs[2][64 * BSTR];
    int rowBase = blockIdx.y * 64;
    int colBase = blockIdx.x * 64;
    if (MODE == 2) {
        // z half only needed for rows with l >= 928
        if (colBase >= DINNER && ((rowBase & (Lseq - 1)) + 64) <= LKEEP) return;
    }
    int tid  = threadIdx.x;
    int wave = tid >> 5;
    int lane = tid & 31;
    int m    = lane & 15;
    int half = lane >> 4;

    v8f acc[4];
#pragma unroll
    for (int j = 0; j < 4; ++j)
#pragma unroll
        for (int r = 0; r < 8; ++r) acc[j][r] = 0.0f;

    // Per-thread staging indices (8 elements of each tile per thread)
    const int ar = tid >> 4, ac = tid & 15;        // A: (ar + 8*t, ac)
    const int br = tid >> 6, bc = tid & 63;        // B: (br + 2*t, bc)

    float ra[8], rb[8];

    // ---- prologue: stage chunk 0 ----
#pragma unroll
    for (int t = 0; t < 8; ++t) {
        int r = ar + 8 * t;
        ra[t] = (ac < K) ? A[(size_t)(rowBase + r) * lda + ac] : 0.0f;
    }
#pragma unroll
    for (int t = 0; t < 8; ++t) {
        int kk = br + 2 * t;
        rb[t] = (kk < K) ? B[(size_t)kk * ldb + colBase + bc] : 0.0f;
    }
#pragma unroll
    for (int t = 0; t < 8; ++t) As[0][(ar + 8 * t) * ASTR + ac] = ra[t];
#pragma unroll
    for (int t = 0; t < 8; ++t) Bs[0][bc * BSTR + (br + 2 * t)] = rb[t];
    __syncthreads();

    for (int k0 = 0; k0 < K; k0 += 16) {
        const int cur = (k0 >> 4) & 1;
        const bool hasNext = (k0 + 16) < K;

        // ---- issue global loads for next chunk (hide latency under WMMA) ----
        if (hasNext) {
            int kn = k0 + 16;
#pragma unroll
            for (int t = 0; t < 8; ++t) {
                int kk = kn + ac;
                ra[t] = (kk < K)
                    ? A[(size_t)(rowBase + ar + 8 * t) * lda + kk] : 0.0f;
            }
#pragma unroll
            for (int t = 0; t < 8; ++t) {
                int kk = kn + br + 2 * t;
                rb[t] = (kk < K) ? B[(size_t)kk * ldb + colBase + bc] : 0.0f;
            }
        }

        // ---- compute: 16 WMMAs on current LDS buffer ----
        const float* Ab = &As[cur][(wave * 16 + m) * ASTR];
#pragma unroll
        for (int kk = 0; kk < 16; kk += 4) {
            v2f a = *(const v2f*)(Ab + kk + 2 * half);
#pragma unroll
            for (int j = 0; j < 4; ++j) {
                v2f bb = *(const v2f*)(&Bs[cur][(j * 16 + m) * BSTR + kk + 2 * half]);
                acc[j] = __builtin_amdgcn_wmma_f32_16x16x4_f32(
                    false, a, false, bb, (short)0, acc[j], false, false);
            }
        }

        // ---- drain staged regs into the other LDS buffer ----
        if (hasNext) {
#pragma unroll
            for (int t = 0; t < 8; ++t)
                As[cur ^ 1][(ar + 8 * t) * ASTR + ac] = ra[t];
#pragma unroll
            for (int t = 0; t < 8; ++t)
                Bs[cur ^ 1][bc * BSTR + (br + 2 * t)] = rb[t];
        }
        __syncthreads();
    }

    // Epilogue: C/D layout — VGPR r: lanes 0-15 -> M=r, lanes 16-31 -> M=r+8
#pragma unroll
    for (int j = 0; j < 4; ++j) {
#pragma unroll
        for (int r = 0; r < 8; ++r) {
            int row = rowBase + wave * 16 + half * 8 + r;
            int col = colBase + j * 16 + m;
            float v = acc[j][r];
            if (MODE == 0) {
                C[(size_t)row * ldc + col] = v;
            } else if (MODE == 1) {
                int l = row & (Lseq - 1);
                float freq = __expf(POSC * (float)(col & ~1));
                float ang  = (float)l * freq;
                v += (col & 1) ? __cosf(ang) : __sinf(ang);
                C[(size_t)row * ldc + col] = v;
            } else if (MODE == 2) {
                if (col < DINNER) {
                    C[(size_t)row * DINNER + col] = v;   // u (pre-conv)
                } else {
                    int l = row & (Lseq - 1);
                    if (l >= LKEEP) {
                        int b = row >> 10;
                        aux[(((size_t)(b * PRED + (l - LKEEP))) << 10) +
                            (col - DINNER)] = silu_f(v); // silu(z) gate
                    }
                }
            } else { // MODE 3: dt = softplus(x + bias)
                v += bias[col];
                v = (v > 20.0f) ? v : log1pf(__expf(v));
                C[(size_t)row * ldc + col] = v;
            }
        }
    }
}

// ---------------------------------------------------------------------------
// Depthwise causal conv (k=4) + SiLU:  u[b,l,d]
// ---------------------------------------------------------------------------
__global__ __launch_bounds__(256)
void conv_silu(const float* __restrict__ ur, const float* __restrict__ cw,
               const float* __restrict__ cb, float* __restrict__ u) {
    size_t e = (size_t)blockIdx.x * 256 + threadIdx.x;   // B*L*DINNER elements
    int d   = (int)(e & (DINNER - 1));
    int row = (int)(e >> 10);
    int b = row >> 10, l = row & (Lseq - 1);
    float acc = cb[d];
#pragma unroll
    for (int k = 0; k < DCONV; ++k) {
        int lk = l + k - (DCONV - 1);
        if (lk >= 0)
            acc += ur[(((size_t)((b << 10) + lk)) << 10) + d] * cw[d * DCONV + k];
    }
    u[e] = silu_f(acc);
}

// ---------------------------------------------------------------------------
// Selective scan. Grid (64, 16): 16 channels/block, wave32 -> 2 channels/wave,
// 16 states on a 16-lane half-wave. h update every step; y emitted for the
// last 96 steps, gated and combined with u*D in-place.
// ---------------------------------------------------------------------------
__global__ __launch_bounds__(256)
void scan_kernel(const float* __restrict__ u, const float* __restrict__ dt,
                 const float* __restrict__ xdbl, const float* __restrict__ A_log,
                 const float* __restrict__ Dp, const float* __restrict__ zg,
                 float* __restrict__ yg) {
    int b     = blockIdx.y;
    int dBase = blockIdx.x * 16;
    int tid   = threadIdx.x;
    int wave  = tid >> 5;
    int lane  = tid & 31;
    int s     = lane & 15;
    int half  = lane >> 4;
    int d     = dBase + wave * 2 + half;

    float Aval = -__expf(A_log[d * DSTATE + s]);
    float Dval = Dp[d];
    float h = 0.0f;
    const int rb = b << 10;

    for (int l = 0; l < Lseq; ++l) {
        int row = rb + l;
        float dt_t = dt[((size_t)row << 10) + d];          // broadcast per channel
        float u_t  = u[((size_t)row << 10) + d];
        float Bv   = xdbl[(size_t)row * 64 + DTRANK + s];
        float Cv   = xdbl[(size_t)row * 64 + DTRANK + DSTATE + s];
        h = h * __expf(dt_t * Aval) + (dt_t * u_t) * Bv;
        if (l >= LKEEP) {
            float v = h * Cv;
            v += __shfl_xor(v, 8, 16);
            v += __shfl_xor(v, 4, 16);
            v += __shfl_xor(v, 2, 16);
            v += __shfl_xor(v, 1, 16);
            if (s == 0) {
                size_t idx = (((size_t)(b * PRED + (l - LKEEP))) << 10) + d;
                yg[idx] = (v + u_t * Dval) * zg[idx];
            }
        }
    }
}

// ---------------------------------------------------------------------------
// Head: out[b,t,c] = (mout[row] . W_head[:,c]) * std + mean; one block/row
// ---------------------------------------------------------------------------
__global__ __launch_bounds__(256)
void head_kernel(const float* __restrict__ mout, const float* __restrict__ Wh,
                 const float* __restrict__ meanv, const float* __restrict__ stdv,
                 float* __restrict__ out) {
    int row = blockIdx.x;                 // 0..1535 (= b*96 + t)
    __shared__ float sm[DMODEL];
    __shared__ float part[8][32];
    for (int i = threadIdx.x; i < DMODEL; i += 256)
        sm[i] = mout[(size_t)row * DMODEL + i];
    __syncthreads();
    int c = threadIdx.x & 31;
    int g = threadIdx.x >> 5;             // 8 k-groups of 64
    float p = 0.f;
    if (c < COUT) {
        int k0 = g * 64;
        for (int k = k0; k < k0 + 64; ++k) p += sm[k] * Wh[k * COUT + c];
    }
    part[g][c] = p;
    __syncthreads();
    if (threadIdx.x < COUT) {
        float sum = 0.f;
#pragma unroll
        for (int gg = 0; gg < 8; ++gg) sum += part[gg][threadIdx.x];
        int b = row / PRED;
        int bc = b * COUT + threadIdx.x;
        out[(size_t)row * COUT + threadIdx.x] = sum * stdv[bc] + meanv[bc];
    }
}

// ---------------------------------------------------------------------------
// Launcher
// ---------------------------------------------------------------------------
extern "C" void kernel_launch(void* const* d_in, const int* in_sizes, int n_in,
                              void* d_out, int out_size, void* d_ws, size_t ws_size,
                              hipStream_t stream) {
    const float* x_enc      = (const float*)d_in[0];
    const float* x_mark     = (const float*)d_in[1];
    const float* conv_tok_w = (const float*)d_in[2];
    const float* time_w     = (const float*)d_in[3];
    const float* W_in       = (const float*)d_in[4];
    const float* conv_w     = (const float*)d_in[5];
    const float* conv_b     = (const float*)d_in[6];
    const float* W_xproj    = (const float*)d_in[7];
    const float* W_dt       = (const float*)d_in[8];
    const float* b_dt       = (const float*)d_in[9];
    const float* A_log      = (const float*)d_in[10];
    const float* Dp         = (const float*)d_in[11];
    const float* W_out      = (const float*)d_in[12];
    const float* W_head     = (const float*)d_in[13];

    float* ws    = (float*)d_ws;
    float* meanv = ws;                                       // 512 (336 used)
    float* stdv  = ws + 512;                                 // 512
    float* We    = stdv + 512;                               // 80*512
    float* Xe    = We + (size_t)KEMB * DMODEL;               // 16384*80
    float* emb   = Xe + (size_t)Bsz * Lseq * KEMB;           // 16384*512
    float* u_raw = emb + (size_t)Bsz * Lseq * DMODEL;        // 16384*1024 (reused as dt)
    float* u     = u_raw + (size_t)Bsz * Lseq * DINNER;      // 16384*1024
    float* zg    = u + (size_t)Bsz * Lseq * DINNER;          // 16*96*1024
    float* xdbl  = zg + (size_t)Bsz * PRED * DINNER;         // 16384*64
    float* yg    = xdbl + (size_t)Bsz * Lseq * 64;           // 16*96*1024
    float* mout  = yg + (size_t)Bsz * PRED * DINNER;         // 1536*512
    float* dtbuf = u_raw;                                    // reuse (u_raw dead after conv)
    float* out   = (float*)d_out;
    (void)in_sizes; (void)n_in; (void)out_size; (void)ws_size;

    const int Mfull = Bsz * Lseq;        // 16384
    const int Mpred = Bsz * PRED;        // 1536

    revin_stats<<<Bsz * CIN, 256, 0, stream>>>(x_enc, meanv, stdv);
    build_we<<<(KEMB * DMODEL + 255) / 256, 256, 0, stream>>>(conv_tok_w, time_w, We);
    build_xe<<<(Mfull * KEMB + 255) / 256, 256, 0, stream>>>(x_enc, x_mark, meanv, stdv, Xe);

    // emb = Xe @ We + pos_embedding
    wmma_gemm<1><<<dim3(DMODEL / 64, Mfull / 64), 128, 0, stream>>>(
        Xe, We, emb, nullptr, nullptr, Mfull, DMODEL, KEMB, KEMB, DMODEL, DMODEL);

    // xz = emb @ W_in : u_raw (cols<1024), silu(z) gate for last 96 rows
    wmma_gemm<2><<<dim3(2 * DINNER / 64, Mfull / 64), 128, 0, stream>>>(
        emb, W_in, u_raw, zg, nullptr, Mfull, 2 * DINNER, DMODEL, DMODEL, 2 * DINNER, DINNER);

    conv_silu<<<(Mfull * DINNER) / 256, 256, 0, stream>>>(u_raw, conv_w, conv_b, u);

    // x_dbl = u @ W_xproj
    wmma_gemm<0><<<dim3(64 / 64, Mfull / 64), 128, 0, stream>>>(
        u, W_xproj, xdbl, nullptr, nullptr, Mfull, 64, DINNER, DINNER, 64, 64);

    // dt = softplus(dtr @ W_dt + b_dt)   (dtr = x_dbl[:, :32], lda = 64)
    wmma_gemm<3><<<dim3(DINNER / 64, Mfull / 64), 128, 0, stream>>>(
        xdbl, W_dt, dtbuf, nullptr, b_dt, Mfull, DINNER, DTRANK, 64, DINNER, DINNER);

    scan_kernel<<<dim3(DINNER / 16, Bsz), 256, 0, stream>>>(
        u, dtbuf, xdbl, A_log, Dp, zg, yg);

    // mout = ygated @ W_out (last 96 rows only)
    wmma_gemm<0><<<dim3(DMODEL / 64, Mpred / 64), 128, 0, stream>>>(
        yg, W_out, mout, nullptr, nullptr, Mpred, DMODEL, DINNER, DINNER, DMODEL, DMODEL);

    head_kernel<<<Mpred, 256, 0, stream>>>(mout, W_head, meanv, stdv, out);
}